// AiidkitTEAVGraphEmbedder_50749333570055
// MI455X (gfx1250) — compile-verified
//
#include <hip/hip_runtime.h>
#include <cstdint>

#define D_EA 128
#define N_TABLES 64
#define VOCAB 1024
#define TABLE_ELEMS (N_TABLES * D_EA)   // 8192 floats = 32 KB

typedef float        f32x4 __attribute__((ext_vector_type(4)));
typedef unsigned int u32x4 __attribute__((ext_vector_type(4)));
typedef int          i32x4 __attribute__((ext_vector_type(4)));
typedef int          i32x8 __attribute__((ext_vector_type(8)));

#define LOG2_10000 13.287712379549449f
#define HALF_PI     1.5707963267948966f

// ---------------------------------------------------------------------------
// Stage the 64x128 f32 ent_attr_table (32 KB) into LDS with one TDM descriptor.
// Verified round 1: lowers to `tensor_load_to_lds s[g0], s[g1]` + s_wait_tensorcnt,
// issued once per block (wave 0 only, behind an execz branch).
// ---------------------------------------------------------------------------
static __device__ inline void stage_table(const float* __restrict__ table,
                                          float* s_ea) {
#if __has_builtin(__builtin_amdgcn_tensor_load_to_lds) && \
    __has_builtin(__builtin_amdgcn_s_wait_tensorcnt)
  if ((threadIdx.x >> 5) == 0) {
    const uint64_t ga       = (uint64_t)(uintptr_t)table;
    const uint32_t lds_off  = (uint32_t)(uintptr_t)s_ea;  // flat LDS addr[31:0] = offset
    const uint32_t n_elem   = TABLE_ELEMS;                // 8192 x 4B

    u32x4 g0;
    g0.x = 1u;                                            // count=1, user mode
    g0.y = lds_off;                                       // lds_addr (bytes)
    g0.z = (uint32_t)ga;                                  // global_addr[31:0]
    g0.w = (uint32_t)((ga >> 32) & 0x01FFFFFFu)           // global_addr[56:32]
         | (2u << 30);                                    // type = 2 ("image")

    i32x8 g1;
    g1[0] = (int)(2u << 16);                              // data_size = 2 -> 4 bytes
    g1[1] = (int)((n_elem & 0xFFFFu) << 16);              // tensor_dim0[15:0]
    g1[2] = (int)(1u << 16);                              // tensor_dim0[31:16]=0, tensor_dim1=1
    g1[3] = (int)((n_elem & 0xFFFFu) << 16);              // tensor_dim1[31:16]=0, tile_dim0=8192
    g1[4] = 1;                                            // tile_dim1=1, tile_dim2=0
    g1[5] = (int)n_elem;                                  // tensor_dim0_stride[31:0]
    g1[6] = (int)((n_elem & 0xFFFFu) << 16);              // stride0[47:32]=0, stride1[15:0]
    g1[7] = 0;                                            // stride1[47:16]

    i32x4 gz = {0, 0, 0, 0};
#if defined(__clang_major__) && (__clang_major__ >= 23)
    i32x8 gz8 = {0, 0, 0, 0, 0, 0, 0, 0};
    __builtin_amdgcn_tensor_load_to_lds(g0, g1, gz, gz, gz8, 0);
#else
    __builtin_amdgcn_tensor_load_to_lds(g0, g1, gz, gz, 0);
#endif
    __builtin_amdgcn_s_wait_tensorcnt(0);
  }
#else
  for (int t = threadIdx.x; t < TABLE_ELEMS; t += blockDim.x) s_ea[t] = table[t];
#endif
  __syncthreads();
}

// ---------------------------------------------------------------------------
// cont branch: out[i, 0:128] = ea_table[id[i]][k] + PE_129(day[i], k)
//              out[i, 128]   = val[i]             + PE_129(day[i], 128)
// One wave32 per node; lane owns k in {lane, lane+32, lane+64, lane+96}.
// Output is write-once streaming -> non-temporal stores so the 1 GB stream
// does not evict the L2-resident categ table used by the sibling kernel.
// ---------------------------------------------------------------------------
__global__ void __launch_bounds__(256) cont_kernel(
    const int* __restrict__ ids, const float* __restrict__ vals,
    const int* __restrict__ days, const float* __restrict__ table,
    float* __restrict__ out, int n) {
  __shared__ float s_ea[TABLE_ELEMS];
  stage_table(table, s_ea);

  const int lane = threadIdx.x & 31;
  const int wid  = threadIdx.x >> 5;
  const int wpb  = blockDim.x >> 5;
  const int wg   = blockIdx.x * wpb + wid;
  const int wt   = gridDim.x * wpb;

  const float scale = LOG2_10000 / 129.0f;
  float invdiv[4];
#pragma unroll
  for (int c = 0; c < 4; ++c) {
    const int k = lane + 32 * c;
    invdiv[c] = exp2f(-(float)(k & ~1) * scale);          // 10000^{-floor2(k)/129}
  }
  const float phase     = (lane & 1) ? HALF_PI : 0.0f;    // cos(x) = sin(x + pi/2)
  const float invdiv128 = exp2f(-128.0f * scale);

  for (int i = wg; i < n; i += wt) {
    const float day = (float)__builtin_nontemporal_load(days + i);
    const int   ea  = __builtin_nontemporal_load(ids + i);
    const float* __restrict__ row  = &s_ea[ea * D_EA];
    float*       __restrict__ orow = out + (size_t)i * (D_EA + 1);
#pragma unroll
    for (int c = 0; c < 4; ++c) {
      const int k = lane + 32 * c;
      __builtin_nontemporal_store(row[k] + __sinf(day * invdiv[c] + phase),
                                  orow + k);              // coalesced b32 th:NT
    }
    if (lane == 0)
      __builtin_nontemporal_store(
          __builtin_nontemporal_load(vals + i) + __sinf(day * invdiv128),
          orow + D_EA);                                   // k=128 even -> sin
  }
}

// ---------------------------------------------------------------------------
// categ branch: out[i] = ea_table[id] + categ_tables[id, vocab] + PE_128(day)
// One wave32 per node; lane owns a float4 slice -> 512 B coalesced row bursts.
// The 32 MB categ table stays on the default (RT) cache path so it becomes
// L2-resident (192 MB L2); everything streamed-once is NT; next node's random
// 512 B row is prefetched (global_prefetch_b8).
// ---------------------------------------------------------------------------
__global__ void __launch_bounds__(256) categ_kernel(
    const int* __restrict__ ids, const int* __restrict__ vocs,
    const int* __restrict__ days, const float* __restrict__ table,
    const float* __restrict__ vtab, float* __restrict__ out, int n) {
  __shared__ float s_ea[TABLE_ELEMS];
  stage_table(table, s_ea);

  const int lane = threadIdx.x & 31;
  const int wid  = threadIdx.x >> 5;
  const int wpb  = blockDim.x >> 5;
  const int wg   = blockIdx.x * wpb + wid;
  const int wt   = gridDim.x * wpb;

  const float scale = LOG2_10000 / 128.0f;
  const float inv0  = exp2f(-(float)(4 * lane)     * scale);
  const float inv2  = exp2f(-(float)(4 * lane + 2) * scale);

  for (int i = wg; i < n; i += wt) {
    const int   ea  = __builtin_nontemporal_load(ids + i);
    const int   vv  = __builtin_nontemporal_load(vocs + i);
    const float day = (float)__builtin_nontemporal_load(days + i);

    const int i2 = i + wt;                                // pipeline the random gather
    if (i2 < n) {
      const float* p2 =
          vtab + ((size_t)ids[i2] * VOCAB + vocs[i2]) * D_EA + lane * 4;
      __builtin_prefetch(p2, 0, 3);                       // -> global_prefetch_b8
    }

    const f32x4 v = ((const f32x4*)(vtab + ((size_t)ea * VOCAB + vv) * D_EA))[lane];
    const f32x4 e = ((const f32x4*)(&s_ea[ea * D_EA]))[lane];

    const float a0 = day * inv0;
    const float a2 = day * inv2;
    f32x4 r;
    r.x = v.x + e.x + __sinf(a0);
    r.y = v.y + e.y + __sinf(a0 + HALF_PI);
    r.z = v.z + e.z + __sinf(a2);
    r.w = v.w + e.w + __sinf(a2 + HALF_PI);
    __builtin_nontemporal_store(r, (f32x4*)(out + (size_t)i * D_EA) + lane);
  }
}

extern "C" void kernel_launch(void* const* d_in, const int* in_sizes, int n_in,
                              void* d_out, int out_size, void* d_ws, size_t ws_size,
                              hipStream_t stream) {
  const int n = in_sizes[0];

  const int*   cont_ids   = (const int*)d_in[0];
  const float* cont_vals  = (const float*)d_in[1];
  const int*   cont_days  = (const int*)d_in[2];
  const int*   cat_ids    = (const int*)d_in[3];
  const int*   cat_vocs   = (const int*)d_in[4];
  const int*   cat_days   = (const int*)d_in[5];
  const float* ea_table   = (const float*)d_in[6];
  const float* cat_tables = (const float*)d_in[7];

  float* out_cont = (float*)d_out;
  float* out_cat  = out_cont + (size_t)n * (D_EA + 1);

  dim3 block(256);                                        // 8 wave32 per block
  int grid = (n + 7) / 8;                                 // >= one node per wave
  if (grid > 2048) grid = 2048;
  if (grid < 1) grid = 1;

  cont_kernel<<<grid, block, 0, stream>>>(cont_ids, cont_vals, cont_days,
                                          ea_table, out_cont, n);
  categ_kernel<<<grid, block, 0, stream>>>(cat_ids, cat_vocs, cat_days,
                                           ea_table, cat_tables, out_cat, n);
}